// GAT_karate_63831803953271
// MI455X (gfx1250) — compile-verified
//
#include <hip/hip_runtime.h>
#include <hip/hip_bf16.h>

typedef __attribute__((ext_vector_type(2))) float v2f;
typedef __attribute__((ext_vector_type(8))) float v8f;

#define NEG_SLOPE 0.2f

// ---------------------------------------------------------------------------
// GEMM: H[nrows x FOUT] = X[nrows x FIN] * W[FIN x FOUT] using f32 WMMA 16x16x4
// Block = 128 threads = 4 waves. Each wave computes a 16-row x FOUT tile.
// A-frag layout (ISA 7.12.2, 32-bit A 16x4): lane L holds row (L&15),
//   VGPR0 = K = 2*(L>>4), VGPR1 = K+1.
// B-frag (4x16): lane L holds col (L&15), VGPR0 = K = 2*(L>>4), VGPR1 = K+1.
// C/D (16x16 f32): VGPR i -> row i (lanes 0-15) / row i+8 (lanes 16-31), col lane&15.
// ---------------------------------------------------------------------------
template <int FIN, int FOUT>
__global__ __launch_bounds__(128) void gemm_wmma_kernel(
    const float* __restrict__ X, const float* __restrict__ W,
    float* __restrict__ H, int nrows) {
  __shared__ float lds_x[4][16 * FIN];
  const int wave = threadIdx.x >> 5;
  const int lane = threadIdx.x & 31;
  const int m0 = (blockIdx.x * 4 + wave) * 16;

  // Stage 16 rows x FIN of X into LDS, coalesced float4, rows clamped (keeps EXEC full)
  float* xt = lds_x[wave];
  constexpr int NCHUNK = 16 * FIN / 4;
#pragma unroll 4
  for (int i = lane; i < NCHUNK; i += 32) {
    int r = i / (FIN / 4);
    int c = (i % (FIN / 4)) * 4;
    int row = m0 + r;
    if (row >= nrows) row = nrows - 1;
    float4 v = *(const float4*)(X + (size_t)row * FIN + c);
    *(float4*)(xt + r * FIN + c) = v;
  }
  __syncthreads();

  constexpr int NT = FOUT / 16;
  v8f acc[NT] = {};

  const int arow = lane & 15;
  const int ksel = 2 * (lane >> 4);
  for (int k = 0; k < FIN; k += 4) {
    v2f a;
    a.x = xt[arow * FIN + k + ksel];
    a.y = xt[arow * FIN + k + ksel + 1];
#pragma unroll
    for (int t = 0; t < NT; ++t) {
      int n = t * 16 + (lane & 15);
      v2f b;
      b.x = W[(k + ksel) * FOUT + n];
      b.y = W[(k + ksel + 1) * FOUT + n];
      acc[t] = __builtin_amdgcn_wmma_f32_16x16x4_f32(
          false, a, false, b, (short)0, acc[t], false, false);
    }
  }

  const int row_off = (lane >> 4) * 8;
  const int col = lane & 15;
#pragma unroll
  for (int t = 0; t < NT; ++t) {
#pragma unroll
    for (int i = 0; i < 8; ++i) {
      int row = m0 + row_off + i;
      if (row < nrows) H[(size_t)row * FOUT + t * 16 + col] = acc[t][i];
    }
  }
}

// ---------------------------------------------------------------------------
// Per-node attention scores: s_src[n] = h[n,:].a_src ; s_dst[n] = h[n,:].a_dst
// One wave32 per node, shuffle reduction.
// ---------------------------------------------------------------------------
template <int F>
__global__ __launch_bounds__(256) void node_scores_kernel(
    const float* __restrict__ H, const float* __restrict__ a_src,
    const float* __restrict__ a_dst, float* __restrict__ ssrc,
    float* __restrict__ sdst, int n) {
  int node = blockIdx.x * (blockDim.x >> 5) + (threadIdx.x >> 5);
  int lane = threadIdx.x & 31;
  if (node >= n) return;
  float as = 0.f, ad = 0.f;
#pragma unroll
  for (int f = lane; f < F; f += 32) {
    float h = H[(size_t)node * F + f];
    as += h * a_src[f];
    ad += h * a_dst[f];
  }
#pragma unroll
  for (int off = 16; off > 0; off >>= 1) {
    as += __shfl_down(as, off, 32);
    ad += __shfl_down(ad, off, 32);
  }
  if (lane == 0) {
    ssrc[node] = as;
    sdst[node] = ad;
  }
}

// ordered-float encoding so unsigned atomicMax == float max (handles negatives)
__device__ __forceinline__ unsigned ford(float x) {
  unsigned b = __float_as_uint(x);
  return (b & 0x80000000u) ? ~b : (b | 0x80000000u);
}
__device__ __forceinline__ float funord(unsigned u) {
  unsigned b = (u & 0x80000000u) ? (u ^ 0x80000000u) : ~u;
  return __uint_as_float(b);
}

__device__ __forceinline__ void edge_sd(int i, int E, const int* __restrict__ src,
                                        const int* __restrict__ dst, int& s, int& d) {
  if (i < E) { s = src[i]; d = dst[i]; }
  else       { s = i - E; d = s; }     // appended self loops
}

// Pass 1: e = leaky_relu(s_src[src]+s_dst[dst]); segment max via ordered atomicMax
__global__ __launch_bounds__(256) void edge_pass1_kernel(
    const int* __restrict__ src, const int* __restrict__ dst,
    const float* __restrict__ ssrc, const float* __restrict__ sdst,
    float* __restrict__ ebuf, unsigned* __restrict__ emaxu, int E, int Etot) {
  int i = blockIdx.x * blockDim.x + threadIdx.x;
  if (i >= Etot) return;
  int s, d;
  edge_sd(i, E, src, dst, s, d);
  float e = ssrc[s] + sdst[d];
  e = (e > 0.f) ? e : NEG_SLOPE * e;
  ebuf[i] = e;
  atomicMax(&emaxu[d], ford(e));
}

// Pass 2: ex = exp(e - emax[dst]); segment sum
__global__ __launch_bounds__(256) void edge_pass2_kernel(
    const int* __restrict__ src, const int* __restrict__ dst,
    float* __restrict__ ebuf, const unsigned* __restrict__ emaxu,
    float* __restrict__ denom, int E, int Etot) {
  int i = blockIdx.x * blockDim.x + threadIdx.x;
  if (i >= Etot) return;
  int s, d;
  edge_sd(i, E, src, dst, s, d);
  float ex = __expf(ebuf[i] - funord(emaxu[d]));
  ebuf[i] = ex;
  atomicAdd(&denom[d], ex);
}

// Pass 3: alpha = ex/denom[dst]; out[dst,:] += alpha * h[src,:]
// 16 threads per edge, F/16 features each.
template <int F>
__global__ __launch_bounds__(256) void edge_pass3_kernel(
    const int* __restrict__ src, const int* __restrict__ dst,
    const float* __restrict__ ebuf, const float* __restrict__ denom,
    const float* __restrict__ H, float* __restrict__ alpha_out,
    float* __restrict__ out, int E, int Etot) {
  long long tid = (long long)blockIdx.x * blockDim.x + threadIdx.x;
  long long edge = tid >> 4;
  int fl = (int)(tid & 15);
  if (edge >= Etot) return;
  int s, d;
  edge_sd((int)edge, E, src, dst, s, d);
  float alpha = ebuf[edge] / denom[d];
  if (fl == 0) alpha_out[edge] = alpha;
  constexpr int V = F / 16;
  const float* hp = H + (size_t)s * F + fl * V;
  float* op = out + (size_t)d * F + fl * V;
#pragma unroll
  for (int j = 0; j < V; ++j) atomicAdd(op + j, alpha * hp[j]);
}

// out[n,f] += b[f]; optional ELU
__global__ __launch_bounds__(256) void bias_act_kernel(
    float* __restrict__ H, const float* __restrict__ b, long long total, int F,
    int do_elu) {
  long long i = (long long)blockIdx.x * blockDim.x + threadIdx.x;
  if (i >= total) return;
  int f = (int)(i % F);
  float v = H[i] + b[f];
  if (do_elu) v = (v > 0.f) ? v : (__expf(v) - 1.0f);
  H[i] = v;
}

static inline int cdiv(long long a, long long b) { return (int)((a + b - 1) / b); }

extern "C" void kernel_launch(void* const* d_in, const int* in_sizes, int n_in,
                              void* d_out, int out_size, void* d_ws, size_t ws_size,
                              hipStream_t stream) {
  constexpr int FIN = 128, FH = 64, FO = 32;
  const float* x      = (const float*)d_in[0];
  const int*   eidx   = (const int*)d_in[1];
  const float* W1     = (const float*)d_in[2];
  const float* a_src1 = (const float*)d_in[3];
  const float* a_dst1 = (const float*)d_in[4];
  const float* b1     = (const float*)d_in[5];
  const float* W2     = (const float*)d_in[6];
  const float* a_src2 = (const float*)d_in[7];
  const float* a_dst2 = (const float*)d_in[8];
  const float* b2     = (const float*)d_in[9];

  const int N = in_sizes[0] / FIN;
  const int E = in_sizes[1] / 2;
  const int Etot = E + N;
  const int* src = eidx;       // edge_index[0]
  const int* dst = eidx + E;   // edge_index[1]

  // workspace carve
  float* hfeat1 = (float*)d_ws;                    // N*FH
  float* agg1   = hfeat1 + (size_t)N * FH;         // N*FH (layer-2 input after act)
  float* hfeat2 = agg1 + (size_t)N * FH;           // N*FO
  float* ebuf   = hfeat2 + (size_t)N * FO;         // Etot
  float* ssrc   = ebuf + (size_t)Etot;             // N
  float* sdst   = ssrc + N;                        // N
  unsigned* emaxu = (unsigned*)(sdst + N);         // N
  float* denom  = (float*)(emaxu + N);             // N

  // output carve: h2 | alpha1 | alpha2
  float* h2     = (float*)d_out;
  float* alpha1 = h2 + (size_t)N * FO;
  float* alpha2 = alpha1 + (size_t)Etot;

  dim3 b256(256), b128(128);

  // ---------------- Layer 1 ----------------
  hipMemsetAsync(agg1, 0, (size_t)N * FH * sizeof(float), stream);
  hipMemsetAsync(emaxu, 0, (size_t)N * sizeof(unsigned), stream);  // ford(-inf) = 0x007fffff > 0, 0 is identity
  hipMemsetAsync(denom, 0, (size_t)N * sizeof(float), stream);

  gemm_wmma_kernel<FIN, FH><<<cdiv(N, 64), b128, 0, stream>>>(x, W1, hfeat1, N);
  node_scores_kernel<FH><<<cdiv(N, 8), b256, 0, stream>>>(hfeat1, a_src1, a_dst1, ssrc, sdst, N);
  edge_pass1_kernel<<<cdiv(Etot, 256), b256, 0, stream>>>(src, dst, ssrc, sdst, ebuf, emaxu, E, Etot);
  edge_pass2_kernel<<<cdiv(Etot, 256), b256, 0, stream>>>(src, dst, ebuf, emaxu, denom, E, Etot);
  edge_pass3_kernel<FH><<<cdiv((long long)Etot * 16, 256), b256, 0, stream>>>(
      src, dst, ebuf, denom, hfeat1, alpha1, agg1, E, Etot);
  bias_act_kernel<<<cdiv((long long)N * FH, 256), b256, 0, stream>>>(
      agg1, b1, (long long)N * FH, FH, /*elu=*/1);

  // ---------------- Layer 2 ----------------
  hipMemsetAsync(h2, 0, (size_t)N * FO * sizeof(float), stream);
  hipMemsetAsync(emaxu, 0, (size_t)N * sizeof(unsigned), stream);
  hipMemsetAsync(denom, 0, (size_t)N * sizeof(float), stream);

  gemm_wmma_kernel<FH, FO><<<cdiv(N, 64), b128, 0, stream>>>(agg1, W2, hfeat2, N);
  node_scores_kernel<FO><<<cdiv(N, 8), b256, 0, stream>>>(hfeat2, a_src2, a_dst2, ssrc, sdst, N);
  edge_pass1_kernel<<<cdiv(Etot, 256), b256, 0, stream>>>(src, dst, ssrc, sdst, ebuf, emaxu, E, Etot);
  edge_pass2_kernel<<<cdiv(Etot, 256), b256, 0, stream>>>(src, dst, ebuf, emaxu, denom, E, Etot);
  edge_pass3_kernel<FO><<<cdiv((long long)Etot * 16, 256), b256, 0, stream>>>(
      src, dst, ebuf, denom, hfeat2, alpha2, h2, E, Etot);
  bias_act_kernel<<<cdiv((long long)N * FO, 256), b256, 0, stream>>>(
      h2, b2, (long long)N * FO, FO, /*elu=*/0);
}